// BasicRNNClassifier_61881888801458
// MI455X (gfx1250) — compile-verified
//
#include <hip/hip_runtime.h>
#include <hip/hip_bf16.h>
#include <stdint.h>

#define U_N 11
#define S_N 15
#define T_N 4096
#define F_N 16
#define CHUNK 64                      // timesteps per LDS buffer
#define CHUNK_FLOATS (CHUNK * F_N)    // 1024 floats = 4KB
#define L2E 1.4426950408889634f
#define EPS_C 1e-8f

// ---------------- CDNA5 async global->LDS path ----------------
#if __has_builtin(__builtin_amdgcn_global_load_async_to_lds_b128)
#define HAVE_ASYNC_BUILTIN 1
#endif
#if __has_builtin(__builtin_amdgcn_s_wait_asynccnt)
#define HAVE_WAIT_BUILTIN 1
#endif

// exact parameter types per hipcc diagnostic: int __vector(4) AS1* (src)
typedef int v4i_t __attribute__((__vector_size__(16)));
typedef __attribute__((address_space(1))) v4i_t* g4_t;
typedef __attribute__((address_space(3))) v4i_t* l4_t;

__device__ __forceinline__ void async_copy16(const float* g, float* l) {
#if defined(HAVE_ASYNC_BUILTIN)
  __builtin_amdgcn_global_load_async_to_lds_b128((g4_t)g, (l4_t)l, 0, 0);
#else
  // ISA: global_load_async_to_lds_b128 vdst(lds addr), vaddr, saddr
  unsigned l32 = (unsigned)(uintptr_t)l;
  asm volatile("global_load_async_to_lds_b128 %0, %1, %2"
               :
               : "v"(l32), "v"((unsigned)(uintptr_t)g), "s"((const void*)0)
               : "memory");
#endif
}

#if defined(HAVE_WAIT_BUILTIN)
#define WAIT_ASYNC(n) __builtin_amdgcn_s_wait_asynccnt(n)
#else
#define WAIT_ASYNC(n) asm volatile("s_wait_asynccnt " #n ::: "memory")
#endif

__device__ __forceinline__ float rcp_nr(float d) {
  float r = __builtin_amdgcn_rcpf(d);
  return fmaf(fmaf(-d, r, 1.0f), r, r);   // one Newton step
}

// copy one 64-step (4KB) chunk: 8 async b128 ops, 32 lanes x 16B each
__device__ __forceinline__ void issue_chunk(const float* gin, int c, int lane,
                                            float* ldsbuf) {
  const float* gsrc = gin + (size_t)c * CHUNK_FLOATS;
#pragma unroll
  for (int q = 0; q < 8; ++q) {
    async_copy16(gsrc + q * 128 + lane * 4, ldsbuf + q * 128 + lane * 4);
  }
}

__global__ __launch_bounds__(32, 1)
void ltc_rnn_kernel(const float* __restrict__ inputs,
                    const int*   __restrict__ seq_len,
                    const float* __restrict__ gleak,
                    const float* __restrict__ vleak,
                    const float* __restrict__ cm,
                    const float* __restrict__ sigma,
                    const float* __restrict__ mu,
                    const float* __restrict__ w,
                    const float* __restrict__ erev,
                    const float* __restrict__ smask,
                    const float* __restrict__ s_sigma,
                    const float* __restrict__ s_mu,
                    const float* __restrict__ s_w,
                    const float* __restrict__ s_erev,
                    const float* __restrict__ s_mask,
                    const float* __restrict__ input_w,
                    const float* __restrict__ input_b,
                    const float* __restrict__ output_w,
                    const float* __restrict__ output_b,
                    const float* __restrict__ dense_w,
                    const float* __restrict__ dense_b,
                    float* __restrict__ out)
{
  __shared__ float xbuf[2][CHUNK_FLOATS + F_N];   // +1 pad row for pipelined read

  const int b    = blockIdx.x;
  const int lane = threadIdx.x;        // 0..31, one wave per batch
  const int j    = lane & 15;          // neuron column
  const int h    = lane >> 4;          // which half of the i-reduction
  const int jc   = (j < U_N) ? j : (U_N - 1);
  const bool jv  = (j < U_N);

  // ---- per-lane parameter columns (register resident) ----
  // recurrent: half h covers i = h*6 .. h*6+5  (i=11 is a zero pad)
  float rA[6], rB[6], rwm[6], rwme[6];
#pragma unroll
  for (int ii = 0; ii < 6; ++ii) {
    int  i  = h * 6 + ii;
    bool ok = jv && (i < U_N);
    int  ic = ok ? i : 0;
    float sg = sigma[ic * U_N + jc];
    float m  = mu   [ic * U_N + jc];
    float wm = w    [ic * U_N + jc] * smask[ic * U_N + jc];
    float er = erev [ic * U_N + jc];
    rA[ii]   = ok ? (-sg * L2E)    : 0.0f;   // exp2 arg = fma(v_i, rA, rB)
    rB[ii]   = ok ? ( m * sg * L2E): 0.0f;
    rwm[ii]  = ok ? wm             : 0.0f;
    rwme[ii] = ok ? wm * er        : 0.0f;
  }
  // sensory: half h covers i = h*8 .. h*8+7 (i=15 is the dt column -> pad)
  float sA[8], sB[8], swm[8], swme[8];
#pragma unroll
  for (int ii = 0; ii < 8; ++ii) {
    int  i  = h * 8 + ii;
    bool ok = jv && (i < S_N);
    int  ic = ok ? i : 0;
    float sg = s_sigma[ic * U_N + jc];
    float m  = s_mu   [ic * U_N + jc];
    float wm = s_w    [ic * U_N + jc] * s_mask[ic * U_N + jc];
    float er = s_erev [ic * U_N + jc];
    float iw = input_w[ic];
    float ib = input_b[ic];
    sA[ii]   = ok ? (-iw * sg * L2E)      : 0.0f;  // exp2 arg = fma(x_i, sA, sB)
    sB[ii]   = ok ? ((m - ib) * sg * L2E) : 0.0f;
    swm[ii]  = ok ? wm                    : 0.0f;
    swme[ii] = ok ? wm * er               : 0.0f;
  }
  const float gl  = gleak[jc];
  const float gvl = gl * vleak[jc];
  const float cmj = cm[jc];

  const int sl = seq_len[b];                       // 1..T, only need t < sl
  const float* gin = inputs + (size_t)b * T_N * F_N;
  const int nch = (sl + CHUNK - 1) / CHUNK;

  auto sens_eval = [&](const float* xr, float& a_out, float& d_out_, float& el_out) {
    el_out = xr[15];
    float a0 = 0.0f, d0 = 0.0f;
#pragma unroll
    for (int ii = 0; ii < 8; ++ii) {
      float xin = xr[h * 8 + ii];
      float e = __builtin_amdgcn_exp2f(fmaf(xin, sA[ii], sB[ii]));
      float s = __builtin_amdgcn_rcpf(1.0f + e);
      a0 = fmaf(swme[ii], s, a0);
      d0 = fmaf(swm[ii],  s, d0);
    }
    a0 += __shfl_xor(a0, 16, 32);
    d0 += __shfl_xor(d0, 16, 32);
    a_out = a0; d_out_ = d0;
  };

  issue_chunk(gin, 0, lane, &xbuf[0][0]);

  float v = 0.0f;
  for (int c = 0; c < nch; ++c) {
    if (c + 1 < nch) {                 // prefetch next chunk, then wait for this one
      issue_chunk(gin, c + 1, lane, &xbuf[(c + 1) & 1][0]);
      WAIT_ASYNC(8);                   // <=8 outstanding => chunk c is complete
    } else {
      WAIT_ASYNC(0);
    }
    asm volatile("" ::: "memory");

    const float* xb  = &xbuf[c & 1][0];
    const int    tend = (sl - c * CHUNK < CHUNK) ? (sl - c * CHUNK) : CHUNK;

    float snum, sden, el;
    sens_eval(xb, snum, sden, el);
    for (int tt = 0; tt < tend; ++tt) {
      // compute next step's sensory terms first: independent work the scheduler
      // can overlap with the dependent unfold chain below
      float snum_n, sden_n, el_n;
      sens_eval(xb + (tt + 1) * F_N, snum_n, sden_n, el_n);  // pad row at tt==63: discarded

      float cm_t = cmj * rcp_nr(el * (1.0f / 6.0f));
#pragma unroll
      for (int uu = 0; uu < 6; ++uu) {
        float rn = 0.0f, rd = 0.0f;
#pragma unroll
        for (int ii = 0; ii < 6; ++ii) {
          float vi = __shfl(v, h * 6 + ii, 32);              // ds_bpermute broadcast
          float e  = __builtin_amdgcn_exp2f(fmaf(vi, rA[ii], rB[ii]));
          float s  = __builtin_amdgcn_rcpf(1.0f + e);
          rn = fmaf(rwme[ii], s, rn);
          rd = fmaf(rwm[ii],  s, rd);
        }
        rn += __shfl_xor(rn, 16, 32);                        // merge the two i-halves
        rd += __shfl_xor(rd, 16, 32);
        float numer = fmaf(cm_t, v, gvl) + (snum + rn);
        float denom = cm_t + gl + (sden + rd) + EPS_C;
        v = numer * rcp_nr(denom);
      }
      snum = snum_n; sden = sden_n; el = el_n;
    }
  }

  if (lane == 0) {   // lane 0 holds v[0] == the single motor neuron
    float o = fmaf(v, output_w[0], output_b[0]);
    out[b]  = fmaf(o, dense_w[0], dense_b[0]);
  }
}

extern "C" void kernel_launch(void* const* d_in, const int* in_sizes, int n_in,
                              void* d_out, int out_size, void* d_ws, size_t ws_size,
                              hipStream_t stream) {
  (void)n_in; (void)out_size; (void)d_ws; (void)ws_size;
  const float* inputs  = (const float*)d_in[0];
  const int*   seqlen  = (const int*)  d_in[1];
  const float* gleak   = (const float*)d_in[2];
  const float* vleak   = (const float*)d_in[3];
  const float* cm      = (const float*)d_in[4];
  const float* sigma   = (const float*)d_in[5];
  const float* mu      = (const float*)d_in[6];
  const float* w       = (const float*)d_in[7];
  const float* erev    = (const float*)d_in[8];
  const float* smask   = (const float*)d_in[9];
  const float* s_sigma = (const float*)d_in[10];
  const float* s_mu    = (const float*)d_in[11];
  const float* s_w     = (const float*)d_in[12];
  const float* s_erev  = (const float*)d_in[13];
  const float* s_mask  = (const float*)d_in[14];
  const float* input_w = (const float*)d_in[15];
  const float* input_b = (const float*)d_in[16];
  const float* out_w   = (const float*)d_in[17];
  const float* out_b   = (const float*)d_in[18];
  const float* dense_w = (const float*)d_in[19];
  const float* dense_b = (const float*)d_in[20];

  const int B = in_sizes[1];   // = 256 (seq_lengths has one entry per batch)
  ltc_rnn_kernel<<<B, 32, 0, stream>>>(inputs, seqlen, gleak, vleak, cm,
                                       sigma, mu, w, erev, smask,
                                       s_sigma, s_mu, s_w, s_erev, s_mask,
                                       input_w, input_b, out_w, out_b,
                                       dense_w, dense_b, (float*)d_out);
}